// GAT_pyg_59313498357784
// MI455X (gfx1250) — compile-verified
//
#include <hip/hip_runtime.h>
#include <hip/hip_bf16.h>

typedef __attribute__((ext_vector_type(2))) float v2f;
typedef __attribute__((ext_vector_type(8))) float v8f;

#define D 32
#define SLOPE 0.2f
#define EPS 1e-16f

// ---- monotone float <-> uint map for atomicMax on signed floats ----
__device__ __forceinline__ unsigned f2o(float f) {
    unsigned b = __float_as_uint(f);
    return (b & 0x80000000u) ? ~b : (b | 0x80000000u);
}
__device__ __forceinline__ float o2f(unsigned o) {
    unsigned b = (o & 0x80000000u) ? (o & 0x7FFFFFFFu) : ~o;
    return __uint_as_float(b);
}

__device__ __forceinline__ float lrelu(float x) { return x > 0.f ? x : SLOPE * x; }

// ---- init: m = ordered(-inf) = 0, denom = 0, out accumulator = 0 ----
__global__ __launch_bounds__(256) void gat_init(unsigned* __restrict__ m,
                                                float* __restrict__ denom,
                                                float* __restrict__ out, int N) {
    int i = blockIdx.x * blockDim.x + threadIdx.x;
    if (i < N) { m[i] = 0u; denom[i] = 0.f; }
    if (i < N * D) out[i] = 0.f;
}

// ---- H = X @ W  via V_WMMA_F32_16X16X4_F32, one wave per 16-row tile ----
// A 16x4 f32: lanes 0-15 hold M=0..15; VGPR0={K=0|K=2}, VGPR1={K=1|K=3} (lane[4] picks K half)
// B 4x16 f32: mirrored; C/D 16x16: VGPR r -> M=r+(lane>>4)*8, N=lane&15
__global__ __launch_bounds__(256) void gat_gemm(const float* __restrict__ X,
                                                const float* __restrict__ W,
                                                float* __restrict__ H, int N) {
    const int wave = blockIdx.x * 8 + (threadIdx.x >> 5);
    const int lane = threadIdx.x & 31;
    const int m0 = wave * 16;
    if (m0 >= N) return;                       // wave-uniform: EXEC stays all-1s for WMMA
    const int row   = m0 + (lane & 15);
    const int col   = lane & 15;
    const int khalf = (lane >> 4) * 2;         // 0 or 2
    v8f c0 = {}; v8f c1 = {};
#pragma unroll
    for (int kb = 0; kb < 8; ++kb) {
        const int k0 = kb * 4 + khalf;
        v2f a;  a.x  = X[row * D + k0];        a.y  = X[row * D + k0 + 1];
        v2f b0; b0.x = W[k0 * D + col];        b0.y = W[(k0 + 1) * D + col];
        v2f b1; b1.x = W[k0 * D + col + 16];   b1.y = W[(k0 + 1) * D + col + 16];
        c0 = __builtin_amdgcn_wmma_f32_16x16x4_f32(false, a, false, b0, (short)0, c0, false, false);
        c1 = __builtin_amdgcn_wmma_f32_16x16x4_f32(false, a, false, b1, (short)0, c1, false, false);
    }
    const int rbase = m0 + (lane >> 4) * 8;
#pragma unroll
    for (int r = 0; r < 8; ++r) {
        H[(rbase + r) * D + col]      = c0[r];
        H[(rbase + r) * D + col + 16] = c1[r];
    }
}

// ---- per-node attention dots: as = H@a_src, ad = H@a_dst ----
__global__ __launch_bounds__(256) void gat_dots(const float* __restrict__ H,
                                                const float* __restrict__ a_src,
                                                const float* __restrict__ a_dst,
                                                float* __restrict__ as_, float* __restrict__ ad_,
                                                int N) {
    int i = blockIdx.x * blockDim.x + threadIdx.x;
    if (i >= N) return;
    float s = 0.f, d = 0.f;
#pragma unroll
    for (int k = 0; k < D; ++k) {
        float h = H[i * D + k];
        s += h * a_src[k];
        d += h * a_dst[k];
    }
    as_[i] = s; ad_[i] = d;
}

// ---- pass 1: segment max of leaky_relu(as[src]+ad[dst]) over dst ----
__global__ __launch_bounds__(256) void gat_edge_max(const int* __restrict__ ei, int E, int N,
                                                    const float* __restrict__ as_,
                                                    const float* __restrict__ ad_,
                                                    unsigned* __restrict__ m) {
    int e = blockIdx.x * blockDim.x + threadIdx.x;
    int Etot = E + N;
    if (e >= Etot) return;
    int s = (e < E) ? ei[e]     : (e - E);
    int d = (e < E) ? ei[E + e] : (e - E);
    float lg = lrelu(as_[s] + ad_[d]);
    atomicMax(&m[d], f2o(lg));
}

// ---- pass 2: denom[dst] += exp(logit - m[dst]) (logit recomputed, cheap: hits L2) ----
__global__ __launch_bounds__(256) void gat_edge_sum(const int* __restrict__ ei, int E, int N,
                                                    const float* __restrict__ as_,
                                                    const float* __restrict__ ad_,
                                                    const unsigned* __restrict__ m,
                                                    float* __restrict__ denom) {
    int e = blockIdx.x * blockDim.x + threadIdx.x;
    int Etot = E + N;
    if (e >= Etot) return;
    int s = (e < E) ? ei[e]     : (e - E);
    int d = (e < E) ? ei[E + e] : (e - E);
    float lg = lrelu(as_[s] + ad_[d]);
    float ex = __expf(lg - o2f(m[d]));
    atomicAdd(&denom[d], ex);
}

// ---- pass 3: wave per edge, lane = channel: out[dst][lane] += alpha * H[src][lane] ----
__global__ __launch_bounds__(256) void gat_edge_aggr(const int* __restrict__ ei, int E, int N,
                                                     const float* __restrict__ as_,
                                                     const float* __restrict__ ad_,
                                                     const unsigned* __restrict__ m,
                                                     const float* __restrict__ denom,
                                                     const float* __restrict__ H,
                                                     float* __restrict__ out) {
    int e    = blockIdx.x * 8 + (threadIdx.x >> 5);
    int lane = threadIdx.x & 31;
    int Etot = E + N;
    if (e >= Etot) return;
    int s = (e < E) ? ei[e]     : (e - E);
    int d = (e < E) ? ei[E + e] : (e - E);
    float lg    = lrelu(as_[s] + ad_[d]);
    float ex    = __expf(lg - o2f(m[d]));
    float alpha = ex / (denom[d] + EPS);
    atomicAdd(&out[d * D + lane], alpha * H[s * D + lane]);
}

// ---- bias + ELU, in place ----
__global__ __launch_bounds__(256) void gat_finalize(float* __restrict__ out,
                                                    const float* __restrict__ b, int N) {
    int i = blockIdx.x * blockDim.x + threadIdx.x;
    if (i >= N * D) return;
    float v = out[i] + b[i & (D - 1)];
    out[i] = (v > 0.f) ? v : (__expf(v) - 1.f);
}

static void run_layer(const float* x_in, const int* ei, int E, int N,
                      const float* W, const float* a_src, const float* a_dst, const float* b,
                      float* H, float* as_, float* ad_, float* denom, unsigned* m,
                      float* out, hipStream_t stream) {
    const int Etot = E + N;
    const int nd_blocks   = (N * D + 255) / 256;
    const int node_blocks = (N + 255) / 256;
    const int edge_blocks = (Etot + 255) / 256;
    const int tile_waves  = (N + 15) / 16;
    const int gemm_blocks = (tile_waves + 7) / 8;
    const int aggr_blocks = (Etot + 7) / 8;

    gat_init<<<nd_blocks, 256, 0, stream>>>(m, denom, out, N);
    gat_gemm<<<gemm_blocks, 256, 0, stream>>>(x_in, W, H, N);
    gat_dots<<<node_blocks, 256, 0, stream>>>(H, a_src, a_dst, as_, ad_, N);
    gat_edge_max<<<edge_blocks, 256, 0, stream>>>(ei, E, N, as_, ad_, m);
    gat_edge_sum<<<edge_blocks, 256, 0, stream>>>(ei, E, N, as_, ad_, m, denom);
    gat_edge_aggr<<<aggr_blocks, 256, 0, stream>>>(ei, E, N, as_, ad_, m, denom, H, out);
    gat_finalize<<<nd_blocks, 256, 0, stream>>>(out, b, N);
}

extern "C" void kernel_launch(void* const* d_in, const int* in_sizes, int n_in,
                              void* d_out, int out_size, void* d_ws, size_t ws_size,
                              hipStream_t stream) {
    const float* x      = (const float*)d_in[0];
    const int*   ei     = (const int*)d_in[1];
    const float* W1     = (const float*)d_in[2];
    const float* a_src1 = (const float*)d_in[3];
    const float* a_dst1 = (const float*)d_in[4];
    const float* b1     = (const float*)d_in[5];
    const float* W2     = (const float*)d_in[6];
    const float* a_src2 = (const float*)d_in[7];
    const float* a_dst2 = (const float*)d_in[8];
    const float* b2     = (const float*)d_in[9];

    const int N = in_sizes[0] / D;
    const int E = in_sizes[1] / 2;

    // workspace layout (floats): H[N*D] | as[N] | ad[N] | denom[N] | m[N] (uint)
    float*    H     = (float*)d_ws;
    float*    as_   = H + (size_t)N * D;
    float*    ad_   = as_ + N;
    float*    denom = ad_ + N;
    unsigned* m     = (unsigned*)(denom + N);

    float* xbar = (float*)d_out;            // output 0
    float* z    = xbar + (size_t)N * D;     // output 1

    // layer 1: x -> z
    run_layer(x, ei, E, N, W1, a_src1, a_dst1, b1, H, as_, ad_, denom, m, z, stream);
    // layer 2: z -> xbar  (H reused)
    run_layer(z, ei, E, N, W2, a_src2, a_dst2, b2, H, as_, ad_, denom, m, xbar, stream);
}